// AttentionPoolingAggregator_48696339202465
// MI455X (gfx1250) — compile-verified
//
#include <hip/hip_runtime.h>
#include <math.h>

#define HIDDEN 256
#define LDS_STRIDE 258   // even (8B alignment for ds b64), not a multiple of 64 banks

typedef __attribute__((ext_vector_type(2))) float v2f;
typedef __attribute__((ext_vector_type(8))) float v8f;

// ---------------------------------------------------------------------------
// C[M x 256] = X[M x 256] @ W[256 x 256]^T   (nn.Linear convention, f32 WMMA)
// grid = (ceil(M/128), 4); block = 256 (8 waves). Each block: 128 rows x 64 cols.
// Wave w handles rows [w*16, w*16+16) x 64 cols = four 16x16 WMMA tiles, so one
// A-fragment load feeds 4 WMMAs (32 f32-FLOP per global byte).
// W panel (64 rows x 256 K, ~66KB) staged in LDS via async global->LDS copies.
// ---------------------------------------------------------------------------
__global__ __launch_bounds__(256) void proj_gemm_kernel(
    const float* __restrict__ X, const float* __restrict__ W,
    float* __restrict__ C, int M) {
  __shared__ float WP[64 * LDS_STRIDE];

  const int tid  = threadIdx.x;
  const int wave = tid >> 5;
  const int lane = tid & 31;
  const int half = lane >> 4;   // 0 = lanes 0-15, 1 = lanes 16-31
  const int l16  = lane & 15;
  const int nbase = blockIdx.y * 64;

  // Stage W rows [nbase, nbase+64), all 256 K, straight into LDS with the
  // CDNA5 async copy path (ASYNCcnt-tracked, no VGPR round trip).
  // Generic LDS pointers carry the offset in the low 32 bits.
  const uintptr_t wp_base = (uintptr_t)(&WP[0]);
  for (int i = tid; i < 64 * 64; i += 256) {   // 64 rows x 64 float4 chunks
    const int r  = i >> 6;
    const int c4 = (i & 63) << 2;
    const float* gp = W + (size_t)(nbase + r) * HIDDEN + c4;
    const unsigned lds_off =
        (unsigned)(wp_base + (unsigned)((r * LDS_STRIDE + c4) << 2));
    asm volatile("global_load_async_to_lds_b128 %0, %1, off"
                 :: "v"(lds_off), "v"(gp) : "memory");
  }
  asm volatile("s_wait_asynccnt 0x0" ::: "memory");
  __syncthreads();

  const int mtile = blockIdx.x * 128 + wave * 16;
  int mrow = mtile + l16;
  if (mrow >= M) mrow = M - 1;   // clamp instead of mask: EXEC must stay all-ones
  const float* aptr = X + (size_t)mrow * HIDDEN + (half << 1);

  v8f acc[4] = {};   // N-tiles [nbase + 16*t, +16)
  for (int k0 = 0; k0 < HIDDEN; k0 += 4) {
    // A frag: lane(half,l16) holds X[m=l16, k0+2*half + {0,1}]
    const v2f a = *reinterpret_cast<const v2f*>(aptr + k0);
#pragma unroll
    for (int t = 0; t < 4; ++t) {
      // B frag: lane(half,l16) holds W[n = nbase+16t+l16, k0+2*half + {0,1}]
      const v2f b = *reinterpret_cast<const v2f*>(
          &WP[(t * 16 + l16) * LDS_STRIDE + k0 + (half << 1)]);
      acc[t] = __builtin_amdgcn_wmma_f32_16x16x4_f32(
          false, a, false, b, (short)0, acc[t], false, false);
    }
  }

  // C/D layout: VGPR r -> rows (mtile + r + 8*half), col = l16 within tile.
#pragma unroll
  for (int r = 0; r < 8; ++r) {
    const int m = mtile + r + (half << 3);
    if (m < M) {
      float* cp = C + (size_t)m * HIDDEN + nbase + l16;
#pragma unroll
      for (int t = 0; t < 4; ++t) cp[t * 16] = acc[t][r];
    }
  }
}

// ---------------------------------------------------------------------------
// Zero out / denom, set smax = -inf.
// ---------------------------------------------------------------------------
__global__ void init_kernel(float* __restrict__ out, float* __restrict__ smax,
                            float* __restrict__ denom, int C) {
  const int i = blockIdx.x * 256 + threadIdx.x;
  if (i < C * HIDDEN) out[i] = 0.0f;
  if (i < C) { smax[i] = -__builtin_inff(); denom[i] = 0.0f; }
}

__device__ __forceinline__ void atomicMaxFloat(float* addr, float val) {
  // Monotone-bit-pattern trick: signed max for >=0, unsigned min for <0.
  if (val >= 0.0f) atomicMax(reinterpret_cast<int*>(addr), __float_as_int(val));
  else atomicMin(reinterpret_cast<unsigned int*>(addr), __float_as_uint(val));
}

// ---------------------------------------------------------------------------
// Per-edge score: s = tanh(A[src] + B[dst]) . v ; segment max via atomics.
// One wave per edge; 8 floats per lane.
// ---------------------------------------------------------------------------
__global__ __launch_bounds__(256) void score_kernel(
    const float* __restrict__ A, const float* __restrict__ B,
    const float* __restrict__ v, const int* __restrict__ src,
    const int* __restrict__ dst, float* __restrict__ score,
    float* __restrict__ smax, int nE) {
  const int e = blockIdx.x * 8 + (threadIdx.x >> 5);
  if (e >= nE) return;
  const int lane = threadIdx.x & 31;
  const int s = src[e];
  const int d = dst[e];

  const float4* ap = reinterpret_cast<const float4*>(A + (size_t)s * HIDDEN) + lane * 2;
  const float4* bp = reinterpret_cast<const float4*>(B + (size_t)d * HIDDEN) + lane * 2;
  const float4* vp = reinterpret_cast<const float4*>(v) + lane * 2;

  float acc = 0.0f;
#pragma unroll
  for (int j = 0; j < 2; ++j) {
    const float4 a = ap[j], b = bp[j], vv = vp[j];
    acc += tanhf(a.x + b.x) * vv.x;
    acc += tanhf(a.y + b.y) * vv.y;
    acc += tanhf(a.z + b.z) * vv.z;
    acc += tanhf(a.w + b.w) * vv.w;
  }
#pragma unroll
  for (int off = 16; off > 0; off >>= 1) acc += __shfl_xor(acc, off, 32);

  if (lane == 0) {
    score[e] = acc;
    atomicMaxFloat(&smax[d], acc);
  }
}

// ---------------------------------------------------------------------------
// exp(score - smax[dst]) and segment sum of the exps.
// ---------------------------------------------------------------------------
__global__ void expden_kernel(const float* __restrict__ score,
                              const int* __restrict__ dst,
                              const float* __restrict__ smax,
                              float* __restrict__ expv,
                              float* __restrict__ denom, int nE) {
  const int e = blockIdx.x * 256 + threadIdx.x;
  if (e >= nE) return;
  const int d = dst[e];
  const float ex = expf(score[e] - smax[d]);
  expv[e] = ex;
  atomicAdd(&denom[d], ex);
}

// ---------------------------------------------------------------------------
// out[dst] += news_x[src] * (expv / max(denom,1e-9)). One wave per edge,
// f32 atomics into the L2-resident [C,256] output.
// ---------------------------------------------------------------------------
__global__ __launch_bounds__(256) void scatter_kernel(
    const float* __restrict__ news_x, const int* __restrict__ src,
    const int* __restrict__ dst, const float* __restrict__ expv,
    const float* __restrict__ denom, float* __restrict__ out, int nE) {
  const int e = blockIdx.x * 8 + (threadIdx.x >> 5);
  if (e >= nE) return;
  const int lane = threadIdx.x & 31;
  const int s = src[e];
  const int d = dst[e];
  float den = denom[d];
  den = den > 1e-9f ? den : 1e-9f;
  const float w = expv[e] / den;

  const float4* xp = reinterpret_cast<const float4*>(news_x + (size_t)s * HIDDEN) + lane * 2;
  float* op = out + (size_t)d * HIDDEN + lane * 8;
  const float4 x0 = xp[0];
  const float4 x1 = xp[1];
  atomicAdd(op + 0, x0.x * w);
  atomicAdd(op + 1, x0.y * w);
  atomicAdd(op + 2, x0.z * w);
  atomicAdd(op + 3, x0.w * w);
  atomicAdd(op + 4, x1.x * w);
  atomicAdd(op + 5, x1.y * w);
  atomicAdd(op + 6, x1.z * w);
  atomicAdd(op + 7, x1.w * w);
}

extern "C" void kernel_launch(void* const* d_in, const int* in_sizes, int n_in,
                              void* d_out, int out_size, void* d_ws, size_t ws_size,
                              hipStream_t stream) {
  const float* news_x    = (const float*)d_in[0];
  const float* company_x = (const float*)d_in[1];
  const float* W_news    = (const float*)d_in[2];
  const float* W_company = (const float*)d_in[3];
  const float* v         = (const float*)d_in[4];
  const int*   src       = (const int*)d_in[5];
  const int*   dst       = (const int*)d_in[6];
  float* out = (float*)d_out;

  const int nNews = in_sizes[0] / HIDDEN;   // 200000
  const int C     = in_sizes[1] / HIDDEN;   // 10000
  const int nE    = in_sizes[5];            // 640000

  // Workspace layout (f32): A | B | score | expv | smax | denom  (~219 MB)
  float* A     = (float*)d_ws;
  float* B     = A + (size_t)nNews * HIDDEN;
  float* score = B + (size_t)C * HIDDEN;
  float* expv  = score + nE;
  float* smax  = expv + nE;
  float* denom = smax + C;

  // Node projections via f32 WMMA (factored out of the edge loop).
  dim3 g1((nNews + 127) / 128, 4);
  proj_gemm_kernel<<<g1, 256, 0, stream>>>(news_x, W_news, A, nNews);
  dim3 g2((C + 127) / 128, 4);
  proj_gemm_kernel<<<g2, 256, 0, stream>>>(company_x, W_company, B, C);

  init_kernel<<<C, 256, 0, stream>>>(out, smax, denom, C);

  score_kernel<<<(nE + 7) / 8, 256, 0, stream>>>(A, B, v, src, dst, score, smax, nE);
  expden_kernel<<<(nE + 255) / 256, 256, 0, stream>>>(score, dst, smax, expv, denom, nE);
  scatter_kernel<<<(nE + 7) / 8, 256, 0, stream>>>(news_x, src, dst, expv, denom, out, nE);
}